// SparseGATConv_80762565034629
// MI455X (gfx1250) — compile-verified
//
#include <hip/hip_runtime.h>
#include <hip/hip_bf16.h>
#include <math.h>

// ---------------------------------------------------------------------------
// Sparse GAT on MI455X (gfx1250, wave32).
//   x:[N,128] f32, edges:[2,E] int, W:[8,128,64] f32, a:[8,128,1] f32
//   out:[N, 8*64] f32
// Phase 1: Wh[h] = x @ W[h] via V_WMMA_F32_16X16X4_F32 (exact fp32 WMMA).
// Phase 2: per-node attention projections.
// Phase 3: global per-head max (encoded-uint atomicMax).
// Phase 4: softmax denominator via f32 atomics.
// Phase 5: head-major edge scatter (Wh[h] stays L2-resident: 25.6MB << 192MB).
// ---------------------------------------------------------------------------

typedef __attribute__((ext_vector_type(2))) float v2f;
typedef __attribute__((ext_vector_type(8))) float v8f;

#define HEADS 8
#define IN_F  128
#define OUT_F 64
#define ALPHA 0.2f

// order-preserving float <-> uint encoding for atomicMax
__device__ __forceinline__ unsigned fenc(float v) {
    unsigned u = __float_as_uint(v);
    return (u & 0x80000000u) ? ~u : (u | 0x80000000u);
}
__device__ __forceinline__ float fdec(unsigned u) {
    return (u & 0x80000000u) ? __uint_as_float(u & 0x7FFFFFFFu)
                             : __uint_as_float(~u);
}

// ------------------------- Phase 1: WMMA GEMM ------------------------------
// grid = (ceil(mtiles/8), HEADS), block = 256 (8 waves). One wave -> one
// 16-row x 64-col tile of Wh[h]. K loop: 32 steps of V_WMMA_F32_16X16X4_F32.
__global__ void gat_gemm_wh(const float* __restrict__ x,
                            const float* __restrict__ W,
                            float* __restrict__ Wh, int Nn) {
    const int h     = blockIdx.y;
    const int wave  = threadIdx.x >> 5;
    const int lane  = threadIdx.x & 31;
    const int mtile = blockIdx.x * 8 + wave;
    const int row0  = mtile * 16;
    if (row0 >= Nn) return;                    // wave-uniform: EXEC stays all-1s

    const int lhalf = lane >> 4;               // 0: lanes 0-15, 1: lanes 16-31
    const int lmod  = lane & 15;

    const float* __restrict__ Wp   = W + (size_t)h * (IN_F * OUT_F);
    const float* __restrict__ xrow = x + (size_t)(row0 + lmod) * IN_F + 2 * lhalf;

    v8f c0 = {}; v8f c1 = {}; v8f c2 = {}; v8f c3 = {};

    for (int k = 0; k < IN_F; k += 4) {
        // A 16x4 fp32: VGPR0 = K even base, VGPR1 = K odd; lanes16-31 hold K+2
        v2f a;
        a.x = xrow[k];
        a.y = xrow[k + 1];
        // B 4x16 fp32 per N-subtile, mirrored layout
        const float* wk0 = Wp + (size_t)(k + 2 * lhalf) * OUT_F + lmod;
        const float* wk1 = wk0 + OUT_F;
        v2f b0 = { wk0[0],  wk1[0]  };
        v2f b1 = { wk0[16], wk1[16] };
        v2f b2 = { wk0[32], wk1[32] };
        v2f b3 = { wk0[48], wk1[48] };
        c0 = __builtin_amdgcn_wmma_f32_16x16x4_f32(false, a, false, b0, (short)0, c0, false, false);
        c1 = __builtin_amdgcn_wmma_f32_16x16x4_f32(false, a, false, b1, (short)0, c1, false, false);
        c2 = __builtin_amdgcn_wmma_f32_16x16x4_f32(false, a, false, b2, (short)0, c2, false, false);
        c3 = __builtin_amdgcn_wmma_f32_16x16x4_f32(false, a, false, b3, (short)0, c3, false, false);
    }

    // C/D layout: VGPR r -> row r (lanes 0-15) / row r+8 (lanes 16-31), N=lmod
    float* __restrict__ outp = Wh + ((size_t)h * Nn + row0) * OUT_F;
#pragma unroll
    for (int r = 0; r < 8; ++r) {
        const int m = r + 8 * lhalf;
        float* op = outp + (size_t)m * OUT_F + lmod;
        op[0]  = c0[r];
        op[16] = c1[r];
        op[32] = c2[r];
        op[48] = c3[r];
    }
}

// ----------------- Phase 2: attention projections per node -----------------
__global__ void gat_attproj(const float* __restrict__ Wh,
                            const float* __restrict__ a,
                            float* __restrict__ asrc,
                            float* __restrict__ atgt, int Nn) {
    const int idx = blockIdx.x * blockDim.x + threadIdx.x;
    if (idx >= Nn * HEADS) return;
    const int h = idx / Nn;
    const int n = idx - h * Nn;
    const float* __restrict__ wr = Wh + ((size_t)h * Nn + n) * OUT_F;
    const float* __restrict__ av = a + (size_t)h * (2 * OUT_F);
    float s0 = 0.f, s1 = 0.f;
#pragma unroll
    for (int f = 0; f < OUT_F; f += 4) {
        float4 w = *(const float4*)(wr + f);
        float4 u = *(const float4*)(av + f);
        float4 v = *(const float4*)(av + OUT_F + f);
        s0 += w.x * u.x + w.y * u.y + w.z * u.z + w.w * u.w;
        s1 += w.x * v.x + w.y * v.y + w.z * v.z + w.w * v.w;
    }
    asrc[(size_t)h * Nn + n] = s0;
    atgt[(size_t)h * Nn + n] = s1;
}

// ---------------- Phase 3: global per-head max of e ------------------------
__global__ void gat_edge_max(const int* __restrict__ ed, int E, int Nn,
                             const float* __restrict__ asrc,
                             const float* __restrict__ atgt,
                             unsigned* __restrict__ mxenc) {
    const int e = blockIdx.x * blockDim.x + threadIdx.x;
    float m[HEADS];
#pragma unroll
    for (int h = 0; h < HEADS; ++h) m[h] = -INFINITY;
    if (e < E) {
        const int s = ed[e];
        const int t = ed[E + e];
#pragma unroll
        for (int h = 0; h < HEADS; ++h) {
            const float v  = asrc[(size_t)h * Nn + s] + atgt[(size_t)h * Nn + t];
            const float lr = (v > 0.f) ? v : ALPHA * v;
            m[h] = fmaxf(m[h], -lr);
        }
    }
#pragma unroll
    for (int off = 16; off > 0; off >>= 1) {
#pragma unroll
        for (int h = 0; h < HEADS; ++h)
            m[h] = fmaxf(m[h], __shfl_xor(m[h], off, 32));
    }
    if ((threadIdx.x & 31) == 0) {
#pragma unroll
        for (int h = 0; h < HEADS; ++h) atomicMax(&mxenc[h], fenc(m[h]));
    }
}

// ---------------- Phase 4: softmax denominator -----------------------------
__global__ void gat_edge_denom(const int* __restrict__ ed, int E, int Nn,
                               const float* __restrict__ asrc,
                               const float* __restrict__ atgt,
                               const unsigned* __restrict__ mxenc,
                               float* __restrict__ denom) {
    const int e = blockIdx.x * blockDim.x + threadIdx.x;
    if (e >= E) return;
    const int s = ed[e];
    const int t = ed[E + e];
#pragma unroll
    for (int h = 0; h < HEADS; ++h) {
        const float mx = fdec(mxenc[h]);
        const float v  = asrc[(size_t)h * Nn + s] + atgt[(size_t)h * Nn + t];
        const float lr = (v > 0.f) ? v : ALPHA * v;
        const float ee = __expf(-lr - mx);
        atomicAdd(&denom[(size_t)h * Nn + s], ee);
    }
}

// ---------------- Phase 5: head-major weighted scatter ---------------------
// grid = (ceil(E/8), HEADS), block = 256. One wave per edge; lane covers two
// features. Per-head Wh slice (25.6MB) stays L2-resident.
__global__ void gat_edge_scatter(const int* __restrict__ ed, int E, int Nn,
                                 const float* __restrict__ asrc,
                                 const float* __restrict__ atgt,
                                 const unsigned* __restrict__ mxenc,
                                 const float* __restrict__ denom,
                                 const float* __restrict__ Wh,
                                 float* __restrict__ out) {
    const int h    = blockIdx.y;
    const int wid  = blockIdx.x * (blockDim.x >> 5) + (threadIdx.x >> 5);
    const int lane = threadIdx.x & 31;
    if (wid >= E) return;
    const int s = ed[wid];
    const int t = ed[E + wid];
    const float mx = fdec(mxenc[h]);
    const float v  = asrc[(size_t)h * Nn + s] + atgt[(size_t)h * Nn + t];
    const float lr = (v > 0.f) ? v : ALPHA * v;
    const float ee = __expf(-lr - mx);
    const float att = ee / (denom[(size_t)h * Nn + s] + 1e-10f);

    const float2 wv = *(const float2*)(Wh + ((size_t)h * Nn + t) * OUT_F + 2 * lane);
    float* op = out + (size_t)s * (HEADS * OUT_F) + h * OUT_F + 2 * lane;
    atomicAdd(op,     att * wv.x);
    atomicAdd(op + 1, att * wv.y);
}

// ---------------------------------------------------------------------------
extern "C" void kernel_launch(void* const* d_in, const int* in_sizes, int n_in,
                              void* d_out, int out_size, void* d_ws, size_t ws_size,
                              hipStream_t stream) {
    const float* x  = (const float*)d_in[0];
    const int*   ed = (const int*)d_in[1];
    const float* W  = (const float*)d_in[2];
    const float* a  = (const float*)d_in[3];
    float* out = (float*)d_out;

    const int Nn = in_sizes[0] / IN_F;   // 100000
    const int E  = in_sizes[1] / 2;      // 3200000

    // workspace layout
    float* Wh    = (float*)d_ws;                              // H*Nn*64
    float* asrc  = Wh   + (size_t)HEADS * Nn * OUT_F;         // H*Nn
    float* atgt  = asrc + (size_t)HEADS * Nn;                 // H*Nn
    float* denom = atgt + (size_t)HEADS * Nn;                 // H*Nn
    unsigned* mxenc = (unsigned*)(denom + (size_t)HEADS * Nn);// H

    // deterministic init every call (graph-capture safe async memsets)
    hipMemsetAsync(denom, 0, (size_t)HEADS * Nn * sizeof(float), stream);
    hipMemsetAsync(mxenc, 0, HEADS * sizeof(unsigned), stream);
    hipMemsetAsync(out, 0, (size_t)out_size * sizeof(float), stream);

    // Phase 1: GEMM (WMMA fp32)
    const int mtiles = (Nn + 15) / 16;
    dim3 g1((mtiles + 7) / 8, HEADS);
    gat_gemm_wh<<<g1, 256, 0, stream>>>(x, W, Wh, Nn);

    // Phase 2: projections
    const int tot = Nn * HEADS;
    gat_attproj<<<(tot + 255) / 256, 256, 0, stream>>>(Wh, a, asrc, atgt, Nn);

    // Phase 3: global per-head max
    gat_edge_max<<<(E + 255) / 256, 256, 0, stream>>>(ed, E, Nn, asrc, atgt, mxenc);

    // Phase 4: denominators
    gat_edge_denom<<<(E + 255) / 256, 256, 0, stream>>>(ed, E, Nn, asrc, atgt, mxenc, denom);

    // Phase 5: weighted scatter, head-major for L2 locality
    dim3 g5((E + 7) / 8, HEADS);
    gat_edge_scatter<<<g5, 256, 0, stream>>>(ed, E, Nn, asrc, atgt, mxenc, denom, Wh, out);
}